// DGCNN_32633161515574
// MI455X (gfx1250) — compile-verified
//
#include <hip/hip_runtime.h>

// ---------------- problem constants ----------------
#define BATCH 16
#define NPTS  1024
#define KNEI  40
#define KPAD  48          // 40 padded to 3 WMMA M-tiles (pad rows duplicate neighbor 0)
#define SH    512         // concat feature width (64+64+128+256)
#define Z2    1024

static constexpr float BN_SCALE  = 0.9999950000374997f; // 1/sqrt(1+1e-5)

typedef __attribute__((ext_vector_type(2))) float v2f;  // A/B frag of 16x16x4 f32 WMMA
typedef __attribute__((ext_vector_type(8))) float v8f;  // C/D frag

// exact-f32 MAC tile: D(16x16) += A(16x4) * B(4x16), 2048 FLOP / instruction
__device__ __forceinline__ v8f wmma4(v2f a, v2f b, v8f c) {
  return __builtin_amdgcn_wmma_f32_16x16x4_f32(false, a, false, b, (short)0, c,
                                               false, false);
}

__device__ __forceinline__ float leaky(float z) { return z > 0.f ? z : 0.2f * z; }

// ---------------- kernel 0: per-point squared norms ----------------
template <int C>
__global__ void sqnorm_kernel(const float* __restrict__ x, int stride,
                              float* __restrict__ sq) {
  int p = blockIdx.x * blockDim.x + threadIdx.x;
  if (p >= BATCH * NPTS) return;
  const float* r = x + (size_t)p * stride;
  float s = 0.f;
#pragma unroll
  for (int c = 0; c < C; ++c) s += r[c] * r[c];
  sq[p] = s;
}

// ---------------- kernel 1: fused kNN (WMMA distance strip + iterative top-K) ----
// One WG (4 wave32) = one (batch, 16-row) tile. Distances for all 1024 columns are
// built with V_WMMA_F32_16X16X4_F32 into a 64 KB LDS strip, then each wave extracts
// top-40 of 4 rows via wave-wide arg-max (contiguous b128 LDS rescans, no spills).
template <int C, int STRIDE>
__global__ void __launch_bounds__(128)
knn_kernel(const float* __restrict__ x, const float* __restrict__ sq,
           int* __restrict__ nbr) {
  constexpr int CP = (C + 3) & ~3;
  __shared__ float s_dist[16][NPTS];     // 64 KB (WGP has 320 KB)
  __shared__ float s_row[16][CP];
  __shared__ float s_rsq[16];
  const int tid = threadIdx.x, lane = tid & 31, wv = tid >> 5;
  const int bidx = blockIdx.x >> 6;
  const int r0 = (blockIdx.x & 63) << 4;
  const float* xb = x + (size_t)bidx * NPTS * STRIDE;

  for (int i = tid; i < 16 * CP; i += 128) {
    int m = i / CP, c = i % CP;
    s_row[m][c] = (c < C) ? xb[(size_t)(r0 + m) * STRIDE + c] : 0.f;
  }
  if (tid < 16) s_rsq[tid] = sq[bidx * NPTS + r0 + tid];
  __syncthreads();

  const int hf = lane >> 4, li = lane & 15;
  for (int ct = wv; ct < NPTS / 16; ct += 4) {   // waves split the 64 column tiles
    const int c0 = ct << 4;
    const float* colp = xb + (size_t)(c0 + li) * STRIDE;
    v8f acc = {};
#pragma unroll
    for (int ks = 0; ks < CP / 4; ++ks) {
      const int kb = ks * 4 + hf * 2;
      v2f a, b;
      a.x = s_row[li][kb];
      a.y = s_row[li][kb + 1];
      b.x = (kb < C) ? colp[kb] : 0.f;           // folds away when C%4==0
      b.y = (kb + 1 < C) ? colp[kb + 1] : 0.f;
      acc = wmma4(a, b, acc);
    }
    const float sqc = sq[bidx * NPTS + c0 + li];
#pragma unroll
    for (int r = 0; r < 8; ++r) {
      const int m = r + hf * 8;
      s_dist[m][c0 + li] = 2.f * acc[r] - s_rsq[m] - sqc;  // neg squared distance
    }
  }
  __syncthreads();

  for (int rr = wv; rr < 16; rr += 4) {          // each wave: 4 rows of top-40
    const int gp = bidx * NPTS + r0 + rr;
    float* drow = &s_dist[rr][0];
    for (int it = 0; it < KNEI; ++it) {
      float bv = -__builtin_inff();
      int bc = 0;
#pragma unroll
      for (int j4 = 0; j4 < 8; ++j4) {           // lane owns contiguous 32 cols -> b128
        const float4 q = *(const float4*)(drow + lane * 32 + j4 * 4);
        if (q.x > bv) { bv = q.x; bc = lane * 32 + j4 * 4 + 0; }
        if (q.y > bv) { bv = q.y; bc = lane * 32 + j4 * 4 + 1; }
        if (q.z > bv) { bv = q.z; bc = lane * 32 + j4 * 4 + 2; }
        if (q.w > bv) { bv = q.w; bc = lane * 32 + j4 * 4 + 3; }
      }
#pragma unroll
      for (int off = 16; off >= 1; off >>= 1) {  // wave32 arg-max, low index on ties
        const float ov = __shfl_xor(bv, off, 32);
        const int oc = __shfl_xor(bc, off, 32);
        if (ov > bv || (ov == bv && oc < bc)) { bv = ov; bc = oc; }
      }
      if (lane == 0) nbr[(size_t)gp * KNEI + it] = bc;
      if (lane == (bc >> 5)) drow[bc] = -__builtin_inff();
    }
  }
}

// ---------------- kernel 2: D[p][o] = w_c . center  (WMMA GEMM) ----------------
template <int CIH, int STRIDE, int CO>
__global__ void __launch_bounds__(128)
center_gemm_kernel(const float* __restrict__ x, const float* __restrict__ w,
                   float* __restrict__ D) {
  constexpr int CP = (CIH + 3) & ~3;
  const int lane = threadIdx.x & 31, wv = threadIdx.x >> 5;
  const int p0 = blockIdx.x << 4;
  const int o0 = (blockIdx.y * 4 + wv) << 4;
  const int hf = lane >> 4, li = lane & 15;
  const float* wr = w + (size_t)(o0 + li) * (2 * CIH) + CIH;  // w_c half of w[o][:]
  const float* xp = x + (size_t)(p0 + li) * STRIDE;
  v8f acc = {};
#pragma unroll
  for (int ks = 0; ks < CP / 4; ++ks) {
    const int kb = ks * 4 + hf * 2;
    v2f a, b;
    a.x = (kb < CIH) ? xp[kb] : 0.f;
    a.y = (kb + 1 < CIH) ? xp[kb + 1] : 0.f;
    b.x = (kb < CIH) ? wr[kb] : 0.f;
    b.y = (kb + 1 < CIH) ? wr[kb + 1] : 0.f;
    acc = wmma4(a, b, acc);
  }
#pragma unroll
  for (int r = 0; r < 8; ++r)
    D[(size_t)(p0 + r + hf * 8) * CO + o0 + li] = acc[r];
}

// ---------------- kernel 3: edge conv (gather -> WMMA -> BN/leaky -> max_k) ------
// One wave per point: 48x{CIH} edge features staged in LDS, 3 M-tiles x (CO/16)
// N-tiles x (CIH/4) K-steps of f32 WMMA; accumulators seeded with the precomputed
// w_c.center term, halving the dominant GEMM's K dimension.
template <int CIH, int STRIDE, int CO>
__global__ void __launch_bounds__(128)
edge_conv_kernel(const float* __restrict__ x, const int* __restrict__ nbr,
                 const float* __restrict__ D, const float* __restrict__ w,
                 const float* __restrict__ g, const float* __restrict__ bnb,
                 float* __restrict__ y) {
  constexpr int CP = (CIH + 3) & ~3;
  __shared__ float s_feat[4][KPAD][CP];          // <= 96 KB (layer 4)
  const int lane = threadIdx.x & 31, wv = threadIdx.x >> 5;
  const int p = blockIdx.x * 4 + wv;
  const int bidx = p >> 10;
  const float* xb = x + (size_t)bidx * NPTS * STRIDE;
  const float* cen = x + (size_t)p * STRIDE;
  for (int kk = 0; kk < KPAD; ++kk) {            // pad rows replicate neighbor 0
    const int j = nbr[(size_t)p * KNEI + (kk < KNEI ? kk : 0)];
    const float* nr = xb + (size_t)j * STRIDE;
    for (int c = lane; c < CP; c += 32)
      s_feat[wv][kk][c] = (c < CIH) ? (nr[c] - cen[c]) : 0.f;
  }
  __syncthreads();

  const int hf = lane >> 4, li = lane & 15;
  for (int o0 = 0; o0 < CO; o0 += 16) {
    const int o = o0 + li;
    const float dini = D[(size_t)p * CO + o];    // w_c . center (row-invariant)
    v8f c0, c1, c2;
#pragma unroll
    for (int r = 0; r < 8; ++r) { c0[r] = dini; c1[r] = dini; c2[r] = dini; }
    const float* wr = w + (size_t)o * (2 * CIH); // w_n half
#pragma unroll
    for (int ks = 0; ks < CP / 4; ++ks) {
      const int kb = ks * 4 + hf * 2;
      v2f b, a0, a1, a2;
      b.x = (kb < CIH) ? wr[kb] : 0.f;
      b.y = (kb + 1 < CIH) ? wr[kb + 1] : 0.f;
      a0.x = s_feat[wv][li][kb];       a0.y = s_feat[wv][li][kb + 1];
      a1.x = s_feat[wv][16 + li][kb];  a1.y = s_feat[wv][16 + li][kb + 1];
      a2.x = s_feat[wv][32 + li][kb];  a2.y = s_feat[wv][32 + li][kb + 1];
      c0 = wmma4(a0, b, c0);
      c1 = wmma4(a1, b, c1);
      c2 = wmma4(a2, b, c2);
    }
    const float sc = g[o] * BN_SCALE;
    const float bi = bnb[o];
    float m = -__builtin_inff();
#pragma unroll
    for (int r = 0; r < 8; ++r) {
      m = fmaxf(m, leaky(sc * c0[r] + bi));
      m = fmaxf(m, leaky(sc * c1[r] + bi));
      m = fmaxf(m, leaky(sc * c2[r] + bi));
    }
    m = fmaxf(m, __shfl_xor(m, 16, 32));         // combine the two 8-row halves
    if (hf == 0) y[(size_t)p * SH + o] = m;      // write into concat buffer
  }
}

// ---------------- kernel 4: final linear + max over points ----------------------
__global__ void __launch_bounds__(128)
final_kernel(const float* __restrict__ cat, const float* __restrict__ wf,
             const float* __restrict__ bf, float* __restrict__ out) {
  const int lane = threadIdx.x & 31, wv = threadIdx.x >> 5;
  const int bidx = blockIdx.x;
  const int hf = lane >> 4, li = lane & 15;
  const int o = (blockIdx.y * 4 + wv) * 16 + li;
  const float* wr = wf + (size_t)o * SH;
  float vmax = -__builtin_inff();
  for (int rt = 0; rt < NPTS / 16; ++rt) {
    const float* ar = cat + (size_t)(bidx * NPTS + rt * 16 + li) * SH;
    v8f acc = {};
#pragma unroll 8
    for (int ks = 0; ks < SH / 4; ++ks) {
      const int kb = ks * 4 + hf * 2;
      v2f a, b;
      a.x = ar[kb];  a.y = ar[kb + 1];
      b.x = wr[kb];  b.y = wr[kb + 1];
      acc = wmma4(a, b, acc);
    }
#pragma unroll
    for (int r = 0; r < 8; ++r) vmax = fmaxf(vmax, acc[r]);  // running max over rows
  }
  vmax = fmaxf(vmax, __shfl_xor(vmax, 16, 32));
  if (hf == 0) out[bidx * Z2 + o] = vmax + bf[o];            // bias after max (monotone)
}

// ---------------- driver ----------------
template <int CIH, int STRIDE, int CO>
static void run_layer(const float* x, const float* w, const float* g,
                      const float* b, float* yout, int* nbr, float* sq, float* D,
                      hipStream_t stream) {
  sqnorm_kernel<CIH><<<(BATCH * NPTS + 255) / 256, 256, 0, stream>>>(x, STRIDE, sq);
  knn_kernel<CIH, STRIDE><<<BATCH * (NPTS / 16), 128, 0, stream>>>(x, sq, nbr);
  center_gemm_kernel<CIH, STRIDE, CO>
      <<<dim3(BATCH * NPTS / 16, CO / 64), 128, 0, stream>>>(x, w, D);
  edge_conv_kernel<CIH, STRIDE, CO>
      <<<BATCH * NPTS / 4, 128, 0, stream>>>(x, nbr, D, w, g, b, yout);
}

extern "C" void kernel_launch(void* const* d_in, const int* in_sizes, int n_in,
                              void* d_out, int out_size, void* d_ws, size_t ws_size,
                              hipStream_t stream) {
  (void)in_sizes; (void)n_in; (void)out_size; (void)ws_size;
  const float* x  = (const float*)d_in[0];
  const float* w1 = (const float*)d_in[1];
  const float* g1 = (const float*)d_in[2];
  const float* b1 = (const float*)d_in[3];
  const float* w2 = (const float*)d_in[4];
  const float* g2 = (const float*)d_in[5];
  const float* b2 = (const float*)d_in[6];
  const float* w3 = (const float*)d_in[7];
  const float* g3 = (const float*)d_in[8];
  const float* b3 = (const float*)d_in[9];
  const float* w4 = (const float*)d_in[10];
  const float* g4 = (const float*)d_in[11];
  const float* b4 = (const float*)d_in[12];
  const float* wf = (const float*)d_in[13];
  const float* bf = (const float*)d_in[14];

  char* ws = (char*)d_ws;                     // workspace layout (~64 MB used)
  int*   nbr = (int*)ws;                      // [0,   2.5MB) kNN indices
  float* sq  = (float*)(ws + (4u  << 20));    // [4MB, +64KB) squared norms
  float* D   = (float*)(ws + (8u  << 20));    // [8MB, +16MB) w_c.center terms
  float* cat = (float*)(ws + (32u << 20));    // [32MB,+32MB) concat features [B*N][512]

  run_layer<3,   3,  64 >(x,         w1, g1, b1, cat + 0,   nbr, sq, D, stream);
  run_layer<64,  SH, 64 >(cat + 0,   w2, g2, b2, cat + 64,  nbr, sq, D, stream);
  run_layer<64,  SH, 128>(cat + 64,  w3, g3, b3, cat + 128, nbr, sq, D, stream);
  run_layer<128, SH, 256>(cat + 128, w4, g4, b4, cat + 256, nbr, sq, D, stream);
  final_kernel<<<dim3(BATCH, Z2 / 64), 128, 0, stream>>>(cat, wf, bf,
                                                         (float*)d_out);
}